// Attention_53704271069327
// MI455X (gfx1250) — compile-verified
//
#include <hip/hip_runtime.h>
#include <hip/hip_bf16.h>
#include <math.h>

typedef __attribute__((ext_vector_type(2))) float v2f;
typedef __attribute__((ext_vector_type(8))) float v8f;

#define BSZ 256
#define TSZ 1024
#define RNN_DIM 1024
#define EMB_DIM 512
#define ATT_DIM 128
#define N_FILT 32
#define KSIZE 31
#define LDSPAD 33   // 16x32 conv tile padded; gcd(33,64)=1 -> conflict-free column reads

// ---------------------------------------------------------------------------
// Kernel 1: pq[b,a] = sum_k query[b,k] * w_query[a,k]
// One wave (32 threads) per 16x16 output tile, V_WMMA_F32_16X16X4_F32 over K.
// A(16x4) layout: lane l holds M=l%16, K = 2*(l/16)+{0,1}. B(4x16) mirrored on N.
// C/D: VGPR i -> M = i + 8*(l/16), N = l%16.
// ---------------------------------------------------------------------------
__global__ __launch_bounds__(32) void pq_wmma_kernel(
    const float* __restrict__ query,   // (B, RNN_DIM)
    const float* __restrict__ wq,      // (ATT_DIM, RNN_DIM)
    float* __restrict__ pq)            // (B, ATT_DIM)
{
  const int a0 = blockIdx.x * 16;      // att-dim tile
  const int m0 = blockIdx.y * 16;      // batch tile
  const int lane = threadIdx.x;
  const int mr = lane & 15;
  const int kh = (lane >> 4) << 1;     // 0 or 2

  v8f c = {};
  const float* qrow = query + (size_t)(m0 + mr) * RNN_DIM + kh;
  const float* wrow = wq + (size_t)(a0 + mr) * RNN_DIM + kh;
  for (int k = 0; k < RNN_DIM; k += 4) {
    v2f a, b;
    a.x = qrow[k];  a.y = qrow[k + 1];
    b.x = wrow[k];  b.y = wrow[k + 1];
    c = __builtin_amdgcn_wmma_f32_16x16x4_f32(false, a, false, b,
                                              (short)0, c, false, false);
  }
  const int thalf = (lane >> 4) << 3;
#pragma unroll
  for (int i = 0; i < 8; ++i)
    pq[(size_t)(m0 + i + thalf) * ATT_DIM + a0 + mr] = c[i];
}

// ---------------------------------------------------------------------------
// Kernel 2 (fused): conv1d(aw_cat) -> 16x32 tile in LDS -> WMMA (16x32)x(32x128)
// -> + pq + processed_memory -> tanh -> * v -> reduce over a -> energies(b,t).
// Grid: (T/16, B), block 128 threads = 4 waves, each wave owns 2 N-tiles.
// ---------------------------------------------------------------------------
__global__ __launch_bounds__(128) void loc_energy_kernel(
    const float* __restrict__ aw,      // (B, 2, T)
    const float* __restrict__ pm,      // (B, T, ATT_DIM)
    const float* __restrict__ pq,      // (B, ATT_DIM)
    const float* __restrict__ wconv,   // (N_FILT, 2, KSIZE)
    const float* __restrict__ wlin,    // (ATT_DIM, N_FILT)
    const float* __restrict__ v,       // (ATT_DIM)
    float* __restrict__ energies)      // (B, T)
{
  __shared__ float s_aw[2][46];             // conv input window (16 + 30)
  __shared__ float s_conv[16 * LDSPAD];     // conv tile [t][f]
  __shared__ float s_pq[ATT_DIM];
  __shared__ float s_v[ATT_DIM];
  __shared__ float s_part[4][16];           // per-wave partial energies

  const int b  = blockIdx.y;
  const int t0 = blockIdx.x * 16;
  const int tid = threadIdx.x;

  // stage input window (zero padded) + per-batch vectors
  if (tid < 92) {
    const int ch = tid / 46, o = tid % 46;
    const int gt = t0 - 15 + o;
    float val = 0.f;
    if (gt >= 0 && gt < TSZ) val = aw[((size_t)b * 2 + ch) * TSZ + gt];
    s_aw[ch][o] = val;
  }
  s_pq[tid] = pq[(size_t)b * ATT_DIM + tid];
  s_v[tid]  = v[tid];
  __syncthreads();

  // conv: 512 outputs / 128 threads
#pragma unroll
  for (int q = 0; q < 4; ++q) {
    const int o = tid + q * 128;
    const int f = o & 31;
    const int tt = o >> 5;
    float acc = 0.f;
#pragma unroll
    for (int ch = 0; ch < 2; ++ch) {
      const float* w = wconv + (size_t)(f * 2 + ch) * KSIZE;
      const float* x = &s_aw[ch][tt];
#pragma unroll
      for (int k = 0; k < KSIZE; ++k) acc = fmaf(x[k], w[k], acc);
    }
    s_conv[tt * LDSPAD + f] = acc;
  }
  __syncthreads();

  const int wave = tid >> 5;
  const int lane = tid & 31;
  const int mr = lane & 15;
  const int kh = (lane >> 4) << 1;
  const int thalf = (lane >> 4) << 3;

  float eSum[8];
#pragma unroll
  for (int i = 0; i < 8; ++i) eSum[i] = 0.f;

  for (int nt = wave * 2; nt < wave * 2 + 2; ++nt) {
    const int a0 = nt * 16;
    v8f c = {};
#pragma unroll
    for (int kb = 0; kb < N_FILT; kb += 4) {
      v2f a, bb;
      a.x = s_conv[mr * LDSPAD + kb + kh];
      a.y = s_conv[mr * LDSPAD + kb + kh + 1];
      const float* wl = wlin + (size_t)(a0 + mr) * N_FILT + kb + kh;
      bb.x = wl[0];  bb.y = wl[1];
      c = __builtin_amdgcn_wmma_f32_16x16x4_f32(false, a, false, bb,
                                                (short)0, c, false, false);
    }
    const int ai = a0 + mr;
    const float pqv = s_pq[ai];
    const float vv  = s_v[ai];
#pragma unroll
    for (int i = 0; i < 8; ++i) {
      const int t = t0 + i + thalf;
      float x = c[i] + pqv + pm[((size_t)b * TSZ + t) * ATT_DIM + ai];
      float e = tanhf(x) * vv;
      // reduce over the 16 lanes of each half-wave (one a-subtile)
      e += __shfl_xor(e, 1, 32);
      e += __shfl_xor(e, 2, 32);
      e += __shfl_xor(e, 4, 32);
      e += __shfl_xor(e, 8, 32);
      eSum[i] += e;   // every lane in a half holds its half's sum
    }
  }
  if (lane == 0 || lane == 16) {
    const int base = (lane >> 4) << 3;
#pragma unroll
    for (int i = 0; i < 8; ++i) s_part[wave][base + i] = eSum[i];
  }
  __syncthreads();
  if (tid < 16) {
    const float e = s_part[0][tid] + s_part[1][tid] +
                    s_part[2][tid] + s_part[3][tid];
    energies[(size_t)b * TSZ + t0 + tid] = e;
  }
}

// ---------------------------------------------------------------------------
// Kernel 3: softmax over T per batch row (+ mask -> -inf)
// ---------------------------------------------------------------------------
__global__ __launch_bounds__(256) void softmax_kernel(
    const float* __restrict__ energies,
    const unsigned char* __restrict__ mask,
    float* __restrict__ wout)
{
  __shared__ float red[256];
  const int b = blockIdx.x, tid = threadIdx.x;
  float e[4];
#pragma unroll
  for (int q = 0; q < 4; ++q) {
    const int t = tid + q * 256;
    float x = energies[(size_t)b * TSZ + t];
    if (mask[(size_t)b * TSZ + t]) x = -INFINITY;
    e[q] = x;
  }
  float m = fmaxf(fmaxf(e[0], e[1]), fmaxf(e[2], e[3]));
  red[tid] = m; __syncthreads();
  for (int s = 128; s > 0; s >>= 1) {
    if (tid < s) red[tid] = fmaxf(red[tid], red[tid + s]);
    __syncthreads();
  }
  m = red[0]; __syncthreads();
  float w[4], sum = 0.f;
#pragma unroll
  for (int q = 0; q < 4; ++q) { w[q] = __expf(e[q] - m); sum += w[q]; }
  red[tid] = sum; __syncthreads();
  for (int s = 128; s > 0; s >>= 1) {
    if (tid < s) red[tid] += red[tid + s];
    __syncthreads();
  }
  const float inv = 1.f / red[0];
#pragma unroll
  for (int q = 0; q < 4; ++q)
    wout[(size_t)b * TSZ + tid + q * 256] = w[q] * inv;
}

// ---------------------------------------------------------------------------
// Kernel 4: context[b,d] = sum_t w[b,t] * memory[b,t,d]  (streams 512 MB)
// ---------------------------------------------------------------------------
__global__ __launch_bounds__(256) void context_kernel(
    const float* __restrict__ wts,     // (B, T)
    const float* __restrict__ memory,  // (B, T, EMB_DIM)
    float* __restrict__ ctx)           // (B, EMB_DIM)
{
  __shared__ float s_w[TSZ];
  const int b = blockIdx.x, tid = threadIdx.x;
#pragma unroll
  for (int q = 0; q < 4; ++q)
    s_w[tid + q * 256] = wts[(size_t)b * TSZ + tid + q * 256];
  __syncthreads();

  const float* mb = memory + (size_t)b * TSZ * EMB_DIM;
  float a0 = 0.f, a1 = 0.f;
  for (int t = 0; t < TSZ; ++t) {
    // speculative prefetch a few rows ahead (global_prefetch_b8)
    __builtin_prefetch(&mb[(size_t)(t + 8) * EMB_DIM + tid], 0, 0);
    const float w = s_w[t];
    a0 = fmaf(w, mb[(size_t)t * EMB_DIM + tid], a0);
    a1 = fmaf(w, mb[(size_t)t * EMB_DIM + tid + 256], a1);
  }
  ctx[(size_t)b * EMB_DIM + tid] = a0;
  ctx[(size_t)b * EMB_DIM + tid + 256] = a1;
}

// ---------------------------------------------------------------------------
extern "C" void kernel_launch(void* const* d_in, const int* in_sizes, int n_in,
                              void* d_out, int out_size, void* d_ws, size_t ws_size,
                              hipStream_t stream) {
  const float* query  = (const float*)d_in[0];   // (B, RNN_DIM)
  const float* memory = (const float*)d_in[1];   // (B, T, EMB_DIM)
  const float* pm     = (const float*)d_in[2];   // (B, T, ATT_DIM)
  const float* aw     = (const float*)d_in[3];   // (B, 2, T)
  const unsigned char* mask = (const unsigned char*)d_in[4];  // (B, T) bool
  const float* wq     = (const float*)d_in[5];   // (ATT_DIM, RNN_DIM)
  const float* wconv  = (const float*)d_in[6];   // (N_FILT, 2, KSIZE)
  const float* wlin   = (const float*)d_in[7];   // (ATT_DIM, N_FILT)
  const float* v      = (const float*)d_in[8];   // (ATT_DIM)

  float* out  = (float*)d_out;
  float* ctx  = out;                              // (B, EMB_DIM) first
  float* wout = out + (size_t)BSZ * EMB_DIM;      // (B, T) second

  float* pqw   = (float*)d_ws;                    // (B, ATT_DIM)
  float* energ = pqw + (size_t)BSZ * ATT_DIM;     // (B, T)

  pq_wmma_kernel<<<dim3(ATT_DIM / 16, BSZ / 16), 32, 0, stream>>>(query, wq, pqw);
  loc_energy_kernel<<<dim3(TSZ / 16, BSZ), 128, 0, stream>>>(
      aw, pm, pqw, wconv, wlin, v, energ);
  softmax_kernel<<<BSZ, 256, 0, stream>>>(energ, mask, wout);
  context_kernel<<<BSZ, 256, 0, stream>>>(wout, memory, ctx);
}